// CogNetDTA_69741678952582
// MI455X (gfx1250) — compile-verified
//
#include <hip/hip_runtime.h>
#include <hip/hip_bf16.h>

// ---------------- model constants ----------------
#define D_     256
#define H_     8
#define C_     32
#define FP_    2048
#define ESM_   1280
#define MEM_   64
#define B_     64
#define L_     256
#define NNODE  28224
#define NEDGE  225792
#define EPS_   1e-5f

typedef __attribute__((ext_vector_type(2))) float v2f;
typedef __attribute__((ext_vector_type(8))) float v8f;

// =======================================================================
// fp32 WMMA GEMM:  C(MxN) = act( A(MxK) @ W(KxN) + bias )
// One wave computes a 16x64 strip of C (4 accumulators) using
// V_WMMA_F32_16X16X4_F32.  Requirements: K % 4 == 0, N % 64 == 0,
// M % 16 == 0 (pad inputs to satisfy).  No predication in the k-loop,
// so EXEC stays all-ones and loads are unguarded b64/b32.
// transB: W stored (N,K) row-major, used as W^T.
// act: 0=none, 1=relu
// Fragment layouts (lane L, hi=L>>4, lo=L&15):
//   A(16x4): a[j] = A[m0+lo][k + 2*hi + j]
//   B(4x16): b[j] = B[k + 2*hi + j][n0+lo]
//   D(16x16): acc[r] -> row (m0 + r + 8*hi), col (n0+lo)
// =======================================================================
__global__ void cog_gemm16_wmma(const float* __restrict__ A,
                                const float* __restrict__ W,
                                const float* __restrict__ bias,
                                float* __restrict__ C,
                                int M, int N, int K, int transB, int act)
{
    int wave = blockIdx.x * (blockDim.x >> 5) + (threadIdx.x >> 5);
    int tilesN = N >> 6;                         // 64 output columns per wave
    int totalTiles = (M >> 4) * tilesN;
    if (wave >= totalTiles) return;              // wave-uniform, EXEC stays full
    int tm = wave / tilesN, tn = wave % tilesN;
    int m0 = tm << 4, n0 = tn << 6;
    int lane = threadIdx.x & 31;
    int hi = lane >> 4, lo = lane & 15;

    const float* Arow = A + (size_t)(m0 + lo) * K + 2 * hi;

    v8f acc[4];
#pragma unroll
    for (int t = 0; t < 4; ++t) acc[t] = (v8f){0.f,0.f,0.f,0.f,0.f,0.f,0.f,0.f};

    if (!transB) {
        const float* Wp = W + n0 + lo;
        for (int k = 0; k < K; k += 4) {
            v2f a = *(const v2f*)(Arow + k);
            int r0 = (k + 2 * hi) * N;
#pragma unroll
            for (int t = 0; t < 4; ++t) {
                v2f b;
                b.x = Wp[r0 + 16 * t];
                b.y = Wp[r0 + N + 16 * t];
                acc[t] = __builtin_amdgcn_wmma_f32_16x16x4_f32(
                             false, a, false, b, (short)0, acc[t], false, false);
            }
        }
    } else {
        for (int k = 0; k < K; k += 4) {
            v2f a = *(const v2f*)(Arow + k);
#pragma unroll
            for (int t = 0; t < 4; ++t) {
                v2f b = *(const v2f*)(W + (size_t)(n0 + 16 * t + lo) * K + k + 2 * hi);
                acc[t] = __builtin_amdgcn_wmma_f32_16x16x4_f32(
                             false, a, false, b, (short)0, acc[t], false, false);
            }
        }
    }

#pragma unroll
    for (int t = 0; t < 4; ++t) {
        int col = n0 + 16 * t + lo;
        float bv = bias ? bias[col] : 0.f;
#pragma unroll
        for (int r = 0; r < 8; ++r) {
            float v = acc[t][r] + bv;
            if (act == 1) v = v > 0.f ? v : 0.f;
            C[(size_t)(m0 + r + 8 * hi) * N + col] = v;
        }
    }
}

// ---------------- small utility kernels ----------------
__global__ void cog_fill(float* p, float v, int n)
{
    for (int i = blockIdx.x * blockDim.x + threadIdx.x; i < n; i += gridDim.x * blockDim.x)
        p[i] = v;
}

__global__ void cog_copy(float* dst, const float* src, int n)
{
    for (int i = blockIdx.x * blockDim.x + threadIdx.x; i < n; i += gridDim.x * blockDim.x)
        dst[i] = src[i];
}

// pad columns: src (rows x sc) -> dst (rows x dc), zeros beyond sc
__global__ void cog_pad_cols(float* dst, const float* src, int rows, int sc, int dc)
{
    int i = blockIdx.x * blockDim.x + threadIdx.x;
    if (i >= rows * dc) return;
    int r = i / dc, c = i - r * dc;
    dst[i] = (c < sc) ? src[r * sc + c] : 0.f;
}

// flat pad: first n_src elements copied, rest zero (row-padding a row-major matrix)
__global__ void cog_pad_flat(float* dst, const float* src, int n_src, int n_total)
{
    int i = blockIdx.x * blockDim.x + threadIdx.x;
    if (i >= n_total) return;
    dst[i] = (i < n_src) ? src[i] : 0.f;
}

// copy a (B_ x srcCols) matrix into dst at column offset (concat helper)
__global__ void cog_copy_cols(float* dst, int dstStride, int colOff,
                              const float* src, int srcCols)
{
    int i = blockIdx.x * blockDim.x + threadIdx.x;
    if (i >= B_ * srcCols) return;
    int r = i / srcCols, c = i - r * srcCols;
    dst[r * dstStride + colOff + c] = src[i];
}

// ---------------- contact-map branch ----------------
__global__ void cog_cm_db(const float* __restrict__ cm,
                          const float* __restrict__ W1, const float* __restrict__ b1,
                          const float* __restrict__ W2, const float* __restrict__ b2,
                          float* __restrict__ sb)
{
    __shared__ float red[256];
    int b = blockIdx.x, t = threadIdx.x;
    const float* img = cm + (size_t)b * L_ * L_;
    float s = 0.f;
    for (int p = t; p < L_ * L_; p += 256) s += img[p];
    red[t] = s; __syncthreads();
    for (int st = 128; st > 0; st >>= 1) { if (t < st) red[t] += red[t + st]; __syncthreads(); }
    if (t == 0) {
        float mean = red[0] * (1.f / (L_ * L_));
        float acc = b2[0];
        for (int j = 0; j < 16; ++j) {
            float h = mean * W1[j] + b1[j];
            h = h > 0.f ? h : 0.f;
            acc += h * W2[j];
        }
        sb[b] = 1.f / (1.f + expf(-acc));
    }
}

__global__ void cog_pvec(const float* v, const float* sb, float* p_vec, int n)
{
    int i = blockIdx.x * blockDim.x + threadIdx.x;
    if (i >= n) return;
    p_vec[i] = v[i] * (1.f + sb[i / D_]);
}

// 3x3 SAME conv (32 out-ch), relu, spatial mean -> pooled[b][c]
__global__ void cog_conv_pool(const float* __restrict__ cm,
                              const float* __restrict__ ker,
                              const float* __restrict__ kb,
                              float* __restrict__ pooled)
{
    __shared__ float kw[32 * 9];
    __shared__ float kbias[32];
    __shared__ float red[256];
    int b = blockIdx.x, t = threadIdx.x;
    if (t < 32 * 9) kw[t] = ker[t];
    if (t < 32) kbias[t] = kb[t];
    __syncthreads();
    const float* img = cm + (size_t)b * L_ * L_;
    float sum[32];
#pragma unroll
    for (int c = 0; c < 32; ++c) sum[c] = 0.f;
    for (int p = t; p < L_ * L_; p += 256) {
        int y = p >> 8, x = p & 255;
        float nb[9];
#pragma unroll
        for (int dy = -1; dy <= 1; ++dy)
#pragma unroll
            for (int dx = -1; dx <= 1; ++dx) {
                int yy = y + dy, xx = x + dx;
                nb[(dy + 1) * 3 + (dx + 1)] =
                    (yy >= 0 && yy < L_ && xx >= 0 && xx < L_) ? img[yy * L_ + xx] : 0.f;
            }
        for (int c = 0; c < 32; ++c) {
            float v = kbias[c];
#pragma unroll
            for (int j = 0; j < 9; ++j) v += kw[c * 9 + j] * nb[j];
            sum[c] += v > 0.f ? v : 0.f;
        }
    }
    for (int c = 0; c < 32; ++c) {
        red[t] = sum[c]; __syncthreads();
        for (int st = 128; st > 0; st >>= 1) { if (t < st) red[t] += red[t + st]; __syncthreads(); }
        if (t == 0) pooled[b * 32 + c] = red[0] * (1.f / (L_ * L_));
        __syncthreads();
    }
}

// ---------------- GATv2 ----------------
__device__ inline void cog_atomicMaxFloat(float* addr, float val)
{
    if (val >= 0.f) atomicMax((int*)addr, __float_as_int(val));
    else            atomicMin((unsigned int*)addr, (unsigned int)__float_as_int(val));
}

__global__ void cog_gat_score(const float* __restrict__ xl, const float* __restrict__ xr,
                              const float* __restrict__ att, const int* __restrict__ ei,
                              float* __restrict__ score, float* __restrict__ m)
{
    int e = blockIdx.x * blockDim.x + threadIdx.x;
    if (e >= NEDGE) return;
    int s = ei[e], t = ei[NEDGE + e];
    const float* ls = xl + (size_t)s * D_;
    const float* rt = xr + (size_t)t * D_;
#pragma unroll
    for (int h = 0; h < H_; ++h) {
        float sc = 0.f;
        for (int c = 0; c < C_; ++c) {
            float v = ls[h * C_ + c] + rt[h * C_ + c];
            v = v > 0.f ? v : 0.2f * v;                // leaky_relu(.,0.2)
            sc += v * att[h * C_ + c];
        }
        score[(size_t)e * H_ + h] = sc;
        cog_atomicMaxFloat(&m[t * H_ + h], sc);
    }
}

__global__ void cog_gat_ex(const int* __restrict__ ei, const float* __restrict__ m,
                           float* __restrict__ score, float* __restrict__ denom)
{
    int idx = blockIdx.x * blockDim.x + threadIdx.x;
    if (idx >= NEDGE * H_) return;
    int e = idx >> 3, h = idx & 7;
    int t = ei[NEDGE + e];
    float ex = expf(score[idx] - m[t * H_ + h]);
    score[idx] = ex;
    atomicAdd(&denom[t * H_ + h], ex);
}

__global__ void cog_gat_agg(const float* __restrict__ xl, const float* __restrict__ score,
                            const float* __restrict__ denom, const int* __restrict__ ei,
                            float* __restrict__ agg)
{
    int idx = blockIdx.x * blockDim.x + threadIdx.x;
    if (idx >= NEDGE * H_) return;
    int e = idx >> 3, h = idx & 7;
    int s = ei[e], t = ei[NEDGE + e];
    float alpha = score[idx] / (denom[t * H_ + h] + 1e-16f);
    const float* ls = xl + (size_t)s * D_ + h * C_;
    float* o = agg + (size_t)t * D_ + h * C_;
    for (int c = 0; c < C_; ++c) atomicAdd(&o[c], alpha * ls[c]);
}

__global__ void cog_gat_finish(float* __restrict__ x_next, const float* __restrict__ bias,
                               float* __restrict__ s_sum, int n)
{
    int i = blockIdx.x * blockDim.x + threadIdx.x;
    if (i >= n) return;
    float v = x_next[i] + bias[i & (D_ - 1)];
    v = v > 0.f ? v : (expf(v) - 1.f);
    x_next[i] = v;
    s_sum[i] += v;
}

__global__ void cog_gather_svec(const float* __restrict__ s_feat,
                                const int* __restrict__ nd, const int* __restrict__ np,
                                const int* __restrict__ ns, float* __restrict__ s_vec)
{
    int b = blockIdx.x;
    int start = 0;
    for (int j = 0; j < b; ++j) start += nd[j] + np[j] + ns[j];
    int idx = start + nd[b] + np[b];
    for (int c = threadIdx.x; c < D_; c += blockDim.x)
        s_vec[b * D_ + c] = s_feat[(size_t)idx * D_ + c];
}

// ---------------- memory attention / heads ----------------
__global__ void cog_softmax64(float* a)   // (64,64) rows, one wave per row
{
    int row = blockIdx.x * (blockDim.x >> 5) + (threadIdx.x >> 5);
    if (row >= B_) return;
    int lane = threadIdx.x & 31;
    float v0 = a[row * 64 + lane], v1 = a[row * 64 + 32 + lane];
    float mx = fmaxf(v0, v1);
    for (int off = 16; off > 0; off >>= 1) mx = fmaxf(mx, __shfl_xor(mx, off, 32));
    float e0 = expf(v0 - mx), e1 = expf(v1 - mx);
    float s = e0 + e1;
    for (int off = 16; off > 0; off >>= 1) s += __shfl_xor(s, off, 32);
    a[row * 64 + lane] = e0 / s;
    a[row * 64 + 32 + lane] = e1 / s;
}

__global__ void cog_layernorm256(const float* __restrict__ x, const float* __restrict__ g,
                                 const float* __restrict__ b, float* __restrict__ y)
{
    int row = blockIdx.x * (blockDim.x >> 5) + (threadIdx.x >> 5);
    if (row >= B_) return;
    int lane = threadIdx.x & 31;
    float v[8]; float s = 0.f;
#pragma unroll
    for (int j = 0; j < 8; ++j) { v[j] = x[row * D_ + j * 32 + lane]; s += v[j]; }
    for (int off = 16; off > 0; off >>= 1) s += __shfl_xor(s, off, 32);
    float mu = s * (1.f / D_);
    float var = 0.f;
#pragma unroll
    for (int j = 0; j < 8; ++j) { float d = v[j] - mu; var += d * d; }
    for (int off = 16; off > 0; off >>= 1) var += __shfl_xor(var, off, 32);
    float inv = rsqrtf(var * (1.f / D_) + EPS_);
#pragma unroll
    for (int j = 0; j < 8; ++j) {
        int c = j * 32 + lane;
        y[row * D_ + c] = (v[j] - mu) * inv * g[c] + b[c];
    }
}

__global__ void cog_head_final(const float* __restrict__ hh, const float* __restrict__ g,
                               const float* __restrict__ be, const float* __restrict__ W2,
                               const float* __restrict__ b2, float* __restrict__ out)
{
    int row = blockIdx.x * (blockDim.x >> 5) + (threadIdx.x >> 5);
    if (row >= B_) return;
    int lane = threadIdx.x & 31;
    const float inv_s = rsqrtf(1.f + EPS_);
    float acc = 0.f;
#pragma unroll
    for (int j = 0; j < 8; ++j) {
        int c = j * 32 + lane;
        float v = hh[row * D_ + c] * inv_s * g[c] + be[c];
        v = v > 0.f ? v : 0.2f * v;
        acc += v * W2[c];
    }
    for (int off = 16; off > 0; off >>= 1) acc += __shfl_xor(acc, off, 32);
    if (lane == 0) out[row] = acc + b2[0];
}

__global__ void cog_final_out(const float* attr, const float* repu, float* out)
{
    int i = threadIdx.x;
    if (i < B_) {
        out[i]          = attr[i] - repu[i];
        out[B_ + i]     = attr[i];
        out[2 * B_ + i] = repu[i];
    }
}

// =======================================================================
// host side
// =======================================================================
enum {
    P_ap_W = 0, P_ap_b, P_ds_W1, P_ds_b1, P_ds_W2, P_ds_b2, P_pe_W, P_pe_b,
    P_g1_Wl, P_g1_bl, P_g1_Wr, P_g1_att, P_g1_b,
    P_g2_Wl, P_g2_bl, P_g2_Wr, P_g2_att, P_g2_b,
    P_g3_Wl, P_g3_bl, P_g3_Wr, P_g3_att, P_g3_b,
    P_ce_K, P_ce_bK, P_ce_W, P_ce_b,
    P_dw_Wv, P_dw_bv,
    P_db_W1, P_db_b1, P_db_W2, P_db_b2,
    P_memory, P_mb_W, P_mb_b,
    P_at_W1, P_at_b1, P_at_g, P_at_be, P_at_W2, P_at_b2,
    P_re_W1, P_re_b1, P_re_g, P_re_be, P_re_W2, P_re_b2,
    P_ln_d_g, P_ln_d_b, P_ln_p_g, P_ln_p_b, P_ln_s_g, P_ln_s_b, P_ln_c_g, P_ln_c_b,
    NPARAM
};

static inline void launch_gemm(hipStream_t st, const float* A, const float* W,
                               const float* bias, float* C,
                               int M, int N, int K, int transB, int act)
{
    int tiles = (M / 16) * (N / 64);            // 16x64 strip per wave
    int blocks = (tiles + 7) / 8;               // 8 waves (256 threads) per block
    cog_gemm16_wmma<<<blocks, 256, 0, st>>>(A, W, bias, C, M, N, K, transB, act);
}

extern "C" void kernel_launch(void* const* d_in, const int* in_sizes, int n_in,
                              void* d_out, int out_size, void* d_ws, size_t ws_size,
                              hipStream_t stream)
{
    (void)in_sizes; (void)n_in; (void)out_size; (void)ws_size;

    const float* graph_x     = (const float*)d_in[0];
    const float* drug_seq    = (const float*)d_in[1];
    const float* protein_esm = (const float*)d_in[2];
    const float* cm          = (const float*)d_in[3];
    const float* P[NPARAM];
    for (int k = 0; k < NPARAM; ++k) P[k] = (const float*)d_in[4 + k];
    const int* ei = (const int*)d_in[4 + NPARAM + 0];
    const int* nd = (const int*)d_in[4 + NPARAM + 1];
    const int* np = (const int*)d_in[4 + NPARAM + 2];
    const int* ns = (const int*)d_in[4 + NPARAM + 3];

    float* base = (float*)d_ws;
    size_t off = 0;
    auto alloc = [&](size_t n) { float* p = base + off; off += n; return p; };

    const size_t NC = (size_t)NNODE * D_;
    float* xa     = alloc(NC);            // x_cur (starts as x0)
    float* xb     = alloc(NC);            // x_next / agg
    float* xl     = alloc(NC);
    float* xr     = alloc(NC);
    float* ssum   = alloc(NC);            // x0+h1+h2+h3
    float* xgpad  = alloc((size_t)NNODE * 80);   // graph_x padded 78->80
    float* wappad = alloc(80 * D_);              // ap_W padded 78->80 rows
    float* score  = alloc((size_t)NEDGE * H_);
    float* mbuf   = alloc((size_t)NNODE * H_);
    float* dbuf   = alloc((size_t)NNODE * H_);
    float* hds    = alloc(B_ * 2 * D_);
    float* dvec   = alloc(B_ * D_);
    float* pebuf  = alloc(B_ * D_);
    float* vbuf   = alloc(B_ * D_);
    float* pvecb  = alloc(B_ * D_);
    float* sbbuf  = alloc(B_);
    float* pooled = alloc(B_ * 32);
    float* cvec   = alloc(B_ * D_);
    float* catq   = alloc(B_ * 2 * D_);
    float* qbuf   = alloc(B_ * D_);
    float* attnb  = alloc(B_ * MEM_);
    float* memi   = alloc(B_ * D_);
    float* lnd    = alloc(B_ * D_);
    float* lnp    = alloc(B_ * D_);
    float* lns    = alloc(B_ * D_);
    float* lnc    = alloc(B_ * D_);
    float* cat3a  = alloc(B_ * 3 * D_);
    float* cat3b  = alloc(B_ * 3 * D_);
    float* hha    = alloc(B_ * D_);
    float* hhb    = alloc(B_ * D_);
    float* svec   = alloc(B_ * D_);
    float* attrb  = alloc(B_);
    float* repub  = alloc(B_);

    // ---- drug branch: d_vec ----
    launch_gemm(stream, drug_seq, P[P_ds_W1], P[P_ds_b1], hds, B_, 2 * D_, FP_, 0, 1);
    launch_gemm(stream, hds, P[P_ds_W2], P[P_ds_b2], dvec, B_, D_, 2 * D_, 0, 0);

    // ---- protein branch: p_vec ----
    launch_gemm(stream, protein_esm, P[P_pe_W], P[P_pe_b], pebuf, B_, D_, ESM_, 0, 1);
    launch_gemm(stream, pebuf, P[P_dw_Wv], P[P_dw_bv], vbuf, B_, D_, D_, 0, 0);
    cog_cm_db<<<B_, 256, 0, stream>>>(cm, P[P_db_W1], P[P_db_b1], P[P_db_W2], P[P_db_b2], sbbuf);
    cog_pvec<<<(B_ * D_ + 255) / 256, 256, 0, stream>>>(vbuf, sbbuf, pvecb, B_ * D_);

    // ---- contact-map conv branch: c_vec ----
    cog_conv_pool<<<B_, 256, 0, stream>>>(cm, P[P_ce_K], P[P_ce_bK], pooled);
    launch_gemm(stream, pooled, P[P_ce_W], P[P_ce_b], cvec, B_, D_, 32, 0, 1);

    // ---- graph branch: pad K 78->80, then x0 + 3 GATv2 layers ----
    cog_pad_cols<<<(NNODE * 80 + 255) / 256, 256, 0, stream>>>(xgpad, graph_x, NNODE, 78, 80);
    cog_pad_flat<<<(80 * D_ + 255) / 256, 256, 0, stream>>>(wappad, P[P_ap_W], 78 * D_, 80 * D_);
    launch_gemm(stream, xgpad, wappad, P[P_ap_b], xa, NNODE, D_, 80, 0, 1);
    cog_copy<<<4096, 256, 0, stream>>>(ssum, xa, (int)NC);

    const int WlI[3]  = {P_g1_Wl,  P_g2_Wl,  P_g3_Wl};
    const int blI[3]  = {P_g1_bl,  P_g2_bl,  P_g3_bl};
    const int WrI[3]  = {P_g1_Wr,  P_g2_Wr,  P_g3_Wr};
    const int attI[3] = {P_g1_att, P_g2_att, P_g3_att};
    const int bI[3]   = {P_g1_b,   P_g2_b,   P_g3_b};

    float* xcur = xa;
    float* xnext = xb;
    for (int layer = 0; layer < 3; ++layer) {
        launch_gemm(stream, xcur, P[WlI[layer]], P[blI[layer]], xl, NNODE, D_, D_, 0, 0);
        launch_gemm(stream, xcur, P[WrI[layer]], nullptr,       xr, NNODE, D_, D_, 0, 0);
        cog_fill<<<2048, 256, 0, stream>>>(mbuf, -1e30f, NNODE * H_);
        cog_fill<<<2048, 256, 0, stream>>>(dbuf, 0.f, NNODE * H_);
        cog_fill<<<4096, 256, 0, stream>>>(xnext, 0.f, (int)NC);
        cog_gat_score<<<(NEDGE + 255) / 256, 256, 0, stream>>>(xl, xr, P[attI[layer]], ei, score, mbuf);
        cog_gat_ex<<<(NEDGE * H_ + 255) / 256, 256, 0, stream>>>(ei, mbuf, score, dbuf);
        cog_gat_agg<<<(NEDGE * H_ + 255) / 256, 256, 0, stream>>>(xl, score, dbuf, ei, xnext);
        cog_gat_finish<<<((int)NC + 255) / 256, 256, 0, stream>>>(xnext, P[bI[layer]], ssum, (int)NC);
        float* t = xcur; xcur = xnext; xnext = t;
    }
    cog_gather_svec<<<B_, 256, 0, stream>>>(ssum, nd, np, ns, svec);

    // ---- memory attention ----
    cog_copy_cols<<<(B_ * D_ + 255) / 256, 256, 0, stream>>>(catq, 2 * D_, 0,   dvec,  D_);
    cog_copy_cols<<<(B_ * D_ + 255) / 256, 256, 0, stream>>>(catq, 2 * D_, D_,  pvecb, D_);
    launch_gemm(stream, catq, P[P_mb_W], P[P_mb_b], qbuf, B_, D_, 2 * D_, 0, 0);
    launch_gemm(stream, qbuf, P[P_memory], nullptr, attnb, B_, MEM_, D_, /*transB=*/1, 0);
    cog_softmax64<<<(B_ + 7) / 8, 256, 0, stream>>>(attnb);
    launch_gemm(stream, attnb, P[P_memory], nullptr, memi, B_, D_, MEM_, 0, 0);

    // ---- layernorms ----
    cog_layernorm256<<<(B_ + 7) / 8, 256, 0, stream>>>(dvec,  P[P_ln_d_g], P[P_ln_d_b], lnd);
    cog_layernorm256<<<(B_ + 7) / 8, 256, 0, stream>>>(pvecb, P[P_ln_p_g], P[P_ln_p_b], lnp);
    cog_layernorm256<<<(B_ + 7) / 8, 256, 0, stream>>>(svec,  P[P_ln_s_g], P[P_ln_s_b], lns);
    cog_layernorm256<<<(B_ + 7) / 8, 256, 0, stream>>>(cvec,  P[P_ln_c_g], P[P_ln_c_b], lnc);

    // ---- heads ----
    cog_copy_cols<<<(B_ * D_ + 255) / 256, 256, 0, stream>>>(cat3a, 3 * D_, 0,      lnd, D_);
    cog_copy_cols<<<(B_ * D_ + 255) / 256, 256, 0, stream>>>(cat3a, 3 * D_, D_,     lnp, D_);
    cog_copy_cols<<<(B_ * D_ + 255) / 256, 256, 0, stream>>>(cat3a, 3 * D_, 2 * D_, memi, D_);
    launch_gemm(stream, cat3a, P[P_at_W1], P[P_at_b1], hha, B_, D_, 3 * D_, 0, 0);
    cog_head_final<<<(B_ + 7) / 8, 256, 0, stream>>>(hha, P[P_at_g], P[P_at_be],
                                                     P[P_at_W2], P[P_at_b2], attrb);

    cog_copy_cols<<<(B_ * D_ + 255) / 256, 256, 0, stream>>>(cat3b, 3 * D_, 0,      lns, D_);
    cog_copy_cols<<<(B_ * D_ + 255) / 256, 256, 0, stream>>>(cat3b, 3 * D_, D_,     lnc, D_);
    cog_copy_cols<<<(B_ * D_ + 255) / 256, 256, 0, stream>>>(cat3b, 3 * D_, 2 * D_, memi, D_);
    launch_gemm(stream, cat3b, P[P_re_W1], P[P_re_b1], hhb, B_, D_, 3 * D_, 0, 0);
    cog_head_final<<<(B_ + 7) / 8, 256, 0, stream>>>(hhb, P[P_re_g], P[P_re_be],
                                                     P[P_re_W2], P[P_re_b2], repub);

    cog_final_out<<<1, 64, 0, stream>>>(attrb, repub, (float*)d_out);
}